// GraphAggregator_90426241450484
// MI455X (gfx1250) — compile-verified
//
#include <hip/hip_runtime.h>
#include <math.h>
#include <string.h>

// ---------------- problem constants (match reference) ----------------
#define NNODES 50000
#define DIM    64
#define NEDGES 800000
#define ETOT   (NEDGES + NNODES)   // edges + self loops
#define HEADS1 2
#define SLOPE  0.2f

typedef __attribute__((ext_vector_type(2))) float v2f;
typedef __attribute__((ext_vector_type(8))) float v8f;

// ---------------- ordered-int float max helpers ----------------
__device__ __forceinline__ unsigned f2ord(float f) {
  unsigned u = __float_as_uint(f);
  return (u & 0x80000000u) ? ~u : (u | 0x80000000u);
}
__device__ __forceinline__ float ord2f(unsigned o) {
  unsigned u = (o & 0x80000000u) ? (o & 0x7fffffffu) : ~o;
  return __uint_as_float(u);
}
static unsigned host_f2ord(float f) {
  unsigned u; memcpy(&u, &f, 4);
  return (u & 0x80000000u) ? ~u : (u | 0x80000000u);
}

// ---------------- fill kernels (d_ws/d_out are poisoned) ----------------
__global__ void fill_f32_k(float* __restrict__ p, float v, long n) {
  long i = (long)blockIdx.x * blockDim.x + threadIdx.x;
  if (i < n) p[i] = v;
}
__global__ void fill_u32_k(unsigned* __restrict__ p, unsigned v, long n) {
  long i = (long)blockIdx.x * blockDim.x + threadIdx.x;
  if (i < n) p[i] = v;
}

// ---------------- f32 WMMA GEMM: C[M,Nc] = X[M,K] @ W[K,Nc] ----------------
// One wave32 per 16x16 output tile; K stepped by 4 via V_WMMA_F32_16X16X4_F32.
// Requires M%16==0, K%4==0, Nc%16==0 (true for all calls here).
// A frag (16x4): lane L (half h=L/16): row = L%16, holds K = k0+2h, k0+2h+1.
// B frag (4x16): lane L: col = L%16, VGPR j holds K = k0+2h+j.
// C/D (16x16):   VGPR v, lane L: row = v + 8h, col = L%16.
__global__ void gat_gemm_wmma_f32(const float* __restrict__ X,
                                  const float* __restrict__ W,
                                  float* __restrict__ C,
                                  int M, int K, int Nc) {
  int wave = (blockIdx.x * blockDim.x + threadIdx.x) >> 5;
  int lane = threadIdx.x & 31;
  int ntn  = Nc >> 4;
  int tm   = wave / ntn;
  int tn   = wave % ntn;
  if (tm >= (M >> 4)) return;              // wave-uniform: EXEC stays all-ones
  int row0 = tm << 4, col0 = tn << 4;
  int half = lane >> 4;                    // 0 for lanes 0-15, 1 for 16-31
  int l16  = lane & 15;

  v8f acc = {};
  for (int k0 = 0; k0 < K; k0 += 4) {
    const float* ap = X + (size_t)(row0 + l16) * K + k0 + 2 * half;
    v2f a; a.x = ap[0]; a.y = ap[1];
    const float* bp = W + (size_t)(k0 + 2 * half) * Nc + col0 + l16;
    v2f b; b.x = bp[0]; b.y = bp[Nc];
    acc = __builtin_amdgcn_wmma_f32_16x16x4_f32(
        /*neg_a=*/false, a, /*neg_b=*/false, b,
        /*c_mod=*/(short)0, acc, /*reuse_a=*/false, /*reuse_b=*/false);
  }
  float* cp = C + (size_t)(row0 + 8 * half) * Nc + col0 + l16;
#pragma unroll
  for (int v = 0; v < 8; ++v) cp[(size_t)v * Nc] = acc[v];
}

// ---------------- per-(node,head) attention scores ----------------
__global__ void gat_att_scores(const float* __restrict__ h,
                               const float* __restrict__ att_src,
                               const float* __restrict__ att_dst,
                               float* __restrict__ a_s, float* __restrict__ a_d,
                               int n, int heads) {
  int idx = blockIdx.x * blockDim.x + threadIdx.x;   // (node*heads + head)
  if (idx >= n * heads) return;
  int hh = idx % heads;
  const float* hp = h + (size_t)idx * DIM;
  const float* s  = att_src + hh * DIM;
  const float* d  = att_dst + hh * DIM;
  float va = 0.f, vb = 0.f;
#pragma unroll 8
  for (int i = 0; i < DIM; ++i) { va += hp[i] * s[i]; vb += hp[i] * d[i]; }
  a_s[idx] = va; a_d[idx] = vb;
}

// ---------------- edge pass 1: leaky_relu + segment max ----------------
__global__ void gat_edge_max(const int* __restrict__ ei,
                             const float* __restrict__ a_s,
                             const float* __restrict__ a_d,
                             float* __restrict__ e_out,
                             unsigned* __restrict__ m_ord, int heads) {
  int idx = blockIdx.x * blockDim.x + threadIdx.x;
  int tot = ETOT * heads;
  if (idx >= tot) return;
  int e  = idx / heads, hh = idx % heads;
  int src = (e < NEDGES) ? ei[e]          : (e - NEDGES);
  int dst = (e < NEDGES) ? ei[NEDGES + e] : (e - NEDGES);
  float v = a_s[src * heads + hh] + a_d[dst * heads + hh];
  v = (v > 0.f) ? v : SLOPE * v;          // leaky_relu
  e_out[idx] = v;
  atomicMax(&m_ord[dst * heads + hh], f2ord(v));
}

// ---------------- edge pass 2: exp + weighted scatter aggregation ----------------
// One wave32 per (edge, head); each lane handles 2 of the 64 features.
__global__ void gat_edge_agg(const int* __restrict__ ei,
                             const float* __restrict__ e_val,
                             const unsigned* __restrict__ m_ord,
                             const float* __restrict__ h,
                             float* __restrict__ z,
                             float* __restrict__ acc, int heads) {
  int gtid = blockIdx.x * blockDim.x + threadIdx.x;
  int widx = gtid >> 5;
  int lane = gtid & 31;
  int tot  = ETOT * heads;
  if (widx >= tot) return;
  int e  = widx / heads, hh = widx % heads;
  int src = (e < NEDGES) ? ei[e]          : (e - NEDGES);
  int dst = (e < NEDGES) ? ei[NEDGES + e] : (e - NEDGES);
  float m = ord2f(m_ord[dst * heads + hh]);
  float p = expf(e_val[widx] - m);
  if (lane == 0) atomicAdd(&z[dst * heads + hh], p);
  const float* hs = h   + ((size_t)src * heads + hh) * DIM;
  float*       ap = acc + ((size_t)dst * heads + hh) * DIM;
  atomicAdd(&ap[lane],      hs[lane]      * p);
  atomicAdd(&ap[lane + 32], hs[lane + 32] * p);
}

// ---------------- node finalize: normalize + bias + ELU ----------------
__global__ void gat_finalize(const float* __restrict__ acc,
                             const float* __restrict__ z,
                             const float* __restrict__ bias,
                             float* __restrict__ out, int n, int heads) {
  int idx = blockIdx.x * blockDim.x + threadIdx.x;   // (n*heads+h)*DIM + d
  if (idx >= n * heads * DIM) return;
  int d  = idx % DIM;
  int nh = idx / DIM;
  int hh = nh % heads;
  float v = acc[idx] / (z[nh] + 1e-16f) + bias[hh * DIM + d];
  out[idx] = (v > 0.f) ? v : (expf(v) - 1.f);        // ELU
}

// ---------------- launch ----------------
extern "C" void kernel_launch(void* const* d_in, const int* in_sizes, int n_in,
                              void* d_out, int out_size, void* d_ws, size_t ws_size,
                              hipStream_t stream) {
  (void)in_sizes; (void)n_in; (void)out_size; (void)ws_size;
  const int*   ei   = (const int*)  d_in[0];   // [2, E]
  const float* x0   = (const float*)d_in[1];   // [N, 64]
  const float* W1   = (const float*)d_in[2];   // [64, 128]
  const float* asv1 = (const float*)d_in[3];   // [2, 64]
  const float* adv1 = (const float*)d_in[4];   // [2, 64]
  const float* b1   = (const float*)d_in[5];   // [128]
  const float* W2   = (const float*)d_in[6];   // [128, 64]
  const float* asv2 = (const float*)d_in[7];   // [1, 64]
  const float* adv2 = (const float*)d_in[8];   // [1, 64]
  const float* b2   = (const float*)d_in[9];   // [64]
  float* out = (float*)d_out;                  // [N, 64]

  // ---- workspace partition (floats) ----
  float* ws = (float*)d_ws;
  size_t o = 0;
  float*    h1   = ws + o; o += (size_t)NNODES * HEADS1 * DIM;   // [N,2,64]
  float*    as1  = ws + o; o += (size_t)NNODES * HEADS1;
  float*    ad1  = ws + o; o += (size_t)NNODES * HEADS1;
  unsigned* m1   = (unsigned*)(ws + o); o += (size_t)NNODES * HEADS1;
  float*    z1   = ws + o; o += (size_t)NNODES * HEADS1;
  float*    e1   = ws + o; o += (size_t)ETOT * HEADS1;
  float*    acc1 = ws + o; o += (size_t)NNODES * HEADS1 * DIM;
  float*    x2   = ws + o; o += (size_t)NNODES * HEADS1 * DIM;   // [N,128]
  float*    h2   = ws + o; o += (size_t)NNODES * DIM;            // [N,64]
  float*    as2  = ws + o; o += (size_t)NNODES;
  float*    ad2  = ws + o; o += (size_t)NNODES;
  unsigned* m2   = (unsigned*)(ws + o); o += (size_t)NNODES;
  float*    z2   = ws + o; o += (size_t)NNODES;
  float*    e2   = ws + o; o += (size_t)ETOT;

  const unsigned NEG_ORD = host_f2ord(-1e30f);
  auto blk = [](long n, int t) { return (unsigned)((n + t - 1) / t); };

  // =================== Layer 1 (heads=2) ===================
  // h1 = x0 @ W1  (WMMA f32, 25000 waves)
  {
    long waves = (NNODES / 16) * (128 / 16);
    gat_gemm_wmma_f32<<<blk(waves * 32, 256), 256, 0, stream>>>(x0, W1, h1, NNODES, 64, 128);
  }
  gat_att_scores<<<blk((long)NNODES * HEADS1, 256), 256, 0, stream>>>(h1, asv1, adv1, as1, ad1, NNODES, HEADS1);

  fill_u32_k<<<blk((long)NNODES * HEADS1, 256), 256, 0, stream>>>(m1, NEG_ORD, (long)NNODES * HEADS1);
  gat_edge_max<<<blk((long)ETOT * HEADS1, 256), 256, 0, stream>>>(ei, as1, ad1, e1, m1, HEADS1);

  fill_f32_k<<<blk((long)NNODES * HEADS1, 256), 256, 0, stream>>>(z1, 0.f, (long)NNODES * HEADS1);
  fill_f32_k<<<blk((long)NNODES * HEADS1 * DIM, 256), 256, 0, stream>>>(acc1, 0.f, (long)NNODES * HEADS1 * DIM);
  gat_edge_agg<<<blk((long)ETOT * HEADS1 * 32, 256), 256, 0, stream>>>(ei, e1, m1, h1, z1, acc1, HEADS1);

  gat_finalize<<<blk((long)NNODES * HEADS1 * DIM, 256), 256, 0, stream>>>(acc1, z1, b1, x2, NNODES, HEADS1);

  // =================== Layer 2 (heads=1) ===================
  {
    long waves = (NNODES / 16) * (64 / 16);
    gat_gemm_wmma_f32<<<blk(waves * 32, 256), 256, 0, stream>>>(x2, W2, h2, NNODES, 128, 64);
  }
  gat_att_scores<<<blk((long)NNODES, 256), 256, 0, stream>>>(h2, asv2, adv2, as2, ad2, NNODES, 1);

  fill_u32_k<<<blk((long)NNODES, 256), 256, 0, stream>>>(m2, NEG_ORD, (long)NNODES);
  gat_edge_max<<<blk((long)ETOT, 256), 256, 0, stream>>>(ei, as2, ad2, e2, m2, 1);

  fill_f32_k<<<blk((long)NNODES, 256), 256, 0, stream>>>(z2, 0.f, (long)NNODES);
  fill_f32_k<<<blk((long)NNODES * DIM, 256), 256, 0, stream>>>(out, 0.f, (long)NNODES * DIM);
  gat_edge_agg<<<blk((long)ETOT * 32, 256), 256, 0, stream>>>(ei, e2, m2, h2, z2, out, 1);

  gat_finalize<<<blk((long)NNODES * DIM, 256), 256, 0, stream>>>(out, z2, b2, out, NNODES, 1);
}